// AttentionTransformer_24412594110480
// MI455X (gfx1250) — compile-verified
//
#include <hip/hip_runtime.h>

// TabNet attentive-transformer mask: per-128-row chunk
//   h = a @ W^T  (bias cancels under ghost-BN mean subtraction)
//   ghost BN (mean / biased var over the 128 rows, per column)
//   mask = sparsemax(h_bn * priors) per row over 256 cols
//
// Memory-bound: ~671 MB streamed -> ~29 us floor @ 23.3 TB/s.
// GEMM uses v_wmma_f32_16x16x4_f32 (fp32 in == fp32 reference precision).
// Priors tile (128 KB/WG, the dominant stream) is L2-prefetched during the
// WMMA phase via global_prefetch_b8 so phase 3 reads hit L2.

typedef __attribute__((ext_vector_type(2))) float v2f;
typedef __attribute__((ext_vector_type(4))) float v4f;
typedef __attribute__((ext_vector_type(8))) float v8f;

#define VBS      128
#define DA       128
#define DD       256
#define BN_EPS   1e-5f
#define A_STRIDE 132   // padded words/row for A stage: b64 fragment reads conflict-free
#define H_STRIDE 260   // padded words/row for H stage: half-wave bank shift = 32

__global__ __launch_bounds__(512, 1)
void tabnet_mask_kernel(const float* __restrict__ A,
                        const float* __restrict__ P,
                        const float* __restrict__ W,
                        const float* __restrict__ gamma,
                        const float* __restrict__ beta,
                        float* __restrict__ out)
{
    // 130 KB, time-shared: first the staged A chunk, then normalized H.
    __shared__ float lds[VBS * H_STRIDE];

    const int  tid   = threadIdx.x;
    const int  wave  = tid >> 5;       // 0..15 (wave32)
    const int  lane  = tid & 31;
    const int  lrow  = lane & 15;
    const int  lhalf = lane >> 4;      // 0 | 1
    const long rowbase = (long)blockIdx.x * VBS;

    // ---------- Phase 0: stage A chunk (128x128 f32) into padded LDS ----------
    {
        const v4f* Ag = (const v4f*)(A + rowbase * DA);
        #pragma unroll
        for (int i = 0; i < 8; ++i) {
            int f = tid + i * 512;             // float4 index 0..4095
            int r = f >> 5;                    // 32 float4 per row
            int c = (f & 31) << 2;
            *(v4f*)&lds[r * A_STRIDE + c] = Ag[f];
        }
    }

    // L2-prefetch this chunk's priors tile (128 KB = 1024 x 128B lines,
    // 2 lines per thread) so the phase-3 stream overlaps the WMMA phase.
    {
        const float* pbase = P + rowbase * DD;
        __builtin_prefetch(pbase + (size_t)(2 * tid) * 32, 0, 1);
        __builtin_prefetch(pbase + (size_t)(2 * tid + 1) * 32, 0, 1);
    }
    __syncthreads();

    // ---------- Phase 1: GEMM, wave w owns cols [16w, 16w+16) ----------
    const int    c0   = wave << 4;
    const float* wrow = W + (long)(c0 + lrow) * DA + (lhalf << 1);

    v8f acc[8];
    #pragma unroll
    for (int m = 0; m < 8; ++m) acc[m] = (v8f)(0.0f);

    #pragma unroll 2
    for (int k0 = 0; k0 < DA; k0 += 4) {
        // B frag (4x16): lane l<16 -> W[c0+l][k0..k0+1], lane l+16 -> W[c0+l][k0+2..k0+3]
        v2f bfrag = *(const v2f*)(wrow + k0);
        #pragma unroll
        for (int m = 0; m < 8; ++m) {
            // A frag (16x4): same split along K, rows m*16+lrow
            v2f afrag = *(const v2f*)&lds[((m << 4) + lrow) * A_STRIDE + k0 + (lhalf << 1)];
            acc[m] = __builtin_amdgcn_wmma_f32_16x16x4_f32(
                false, afrag, false, bfrag, (short)0, acc[m], false, false);
        }
    }

    // ---------- Phase 2: ghost-BN stats entirely inside the wave ----------
    float s = 0.0f, sq = 0.0f;
    #pragma unroll
    for (int m = 0; m < 8; ++m)
        #pragma unroll
        for (int v = 0; v < 8; ++v) { float x = acc[m][v]; s += x; sq += x * x; }
    s  += __shfl_xor(s,  16, 32);   // combine the two half-wave row groups
    sq += __shfl_xor(sq, 16, 32);

    const int   col   = c0 + lrow;
    const float mean  = s  * (1.0f / 128.0f);
    const float var   = sq * (1.0f / 128.0f) - mean * mean;
    const float scale = gamma[col] * rsqrtf(var + BN_EPS);
    const float shift = beta[col] - mean * scale;     // bias b cancels exactly here

    __syncthreads();   // everyone done reading A region -> safe to overwrite with H

    #pragma unroll
    for (int m = 0; m < 8; ++m)
        #pragma unroll
        for (int v = 0; v < 8; ++v) {
            int r = (m << 4) + v + (lhalf << 3);
            lds[r * H_STRIDE + col] = acc[m][v] * scale + shift;  // conflict-free
        }
    __syncthreads();

    // ---------- Phase 3: sparsemax, one wave per row (8 vals / lane) ----------
    #pragma unroll 1
    for (int j = 0; j < 8; ++j) {
        const int r = wave + (j << 4);            // rows 0..127 covered once
        float z[8];
        {
            const float* hrow = &lds[r * H_STRIDE + (lane << 3)];
            v4f h0 = *(const v4f*)(hrow);
            v4f h1 = *(const v4f*)(hrow + 4);
            const float* prow = P + (rowbase + r) * DD + (lane << 3);
            v4f p0 = *(const v4f*)(prow);
            v4f p1 = *(const v4f*)(prow + 4);
            z[0] = h0.x * p0.x; z[1] = h0.y * p0.y; z[2] = h0.z * p0.z; z[3] = h0.w * p0.w;
            z[4] = h1.x * p1.x; z[5] = h1.y * p1.y; z[6] = h1.z * p1.z; z[7] = h1.w * p1.w;
        }

        // Michelot fixed point: tau <- (sum_{z>tau} z - 1)/|{z>tau}|, exact at convergence
        float S = 0.0f;
        #pragma unroll
        for (int i = 0; i < 8; ++i) S += z[i];
        #pragma unroll
        for (int off = 16; off; off >>= 1) S += __shfl_xor(S, off, 32);
        float tau = (S - 1.0f) * (1.0f / 256.0f);

        #pragma unroll 1
        for (int it = 0; it < 12; ++it) {
            float Sk = 0.0f, Ck = 0.0f;
            #pragma unroll
            for (int i = 0; i < 8; ++i) {
                if (z[i] > tau) { Sk += z[i]; Ck += 1.0f; }
            }
            #pragma unroll
            for (int off = 16; off; off >>= 1) {
                Sk += __shfl_xor(Sk, off, 32);
                Ck += __shfl_xor(Ck, off, 32);
            }
            tau = (Sk - 1.0f) / fmaxf(Ck, 1.0f);
        }

        v4f o0, o1;
        o0.x = fmaxf(z[0] - tau, 0.0f); o0.y = fmaxf(z[1] - tau, 0.0f);
        o0.z = fmaxf(z[2] - tau, 0.0f); o0.w = fmaxf(z[3] - tau, 0.0f);
        o1.x = fmaxf(z[4] - tau, 0.0f); o1.y = fmaxf(z[5] - tau, 0.0f);
        o1.z = fmaxf(z[6] - tau, 0.0f); o1.w = fmaxf(z[7] - tau, 0.0f);
        float* orow = out + (rowbase + r) * DD + (lane << 3);
        *(v4f*)(orow)     = o0;
        *(v4f*)(orow + 4) = o1;
    }
}

extern "C" void kernel_launch(void* const* d_in, const int* in_sizes, int n_in,
                              void* d_out, int out_size, void* d_ws, size_t ws_size,
                              hipStream_t stream) {
    (void)in_sizes; (void)n_in; (void)d_ws; (void)ws_size; (void)out_size;
    const float* a      = (const float*)d_in[0];
    const float* priors = (const float*)d_in[1];
    const float* W      = (const float*)d_in[2];
    // d_in[3] = bias b: provably no effect on output (cancelled by BN mean subtraction)
    const float* gamma  = (const float*)d_in[4];
    const float* beta   = (const float*)d_in[5];
    float* out = (float*)d_out;

    const int B = 262144;
    dim3 grid(B / VBS);   // 2048 chunks, one per workgroup
    dim3 block(512);      // 16 wave32 waves
    tabnet_mask_kernel<<<grid, block, 0, stream>>>(a, priors, W, gamma, beta, out);
}